// Conv_21208548507745
// MI455X (gfx1250) — compile-verified
//
#include <hip/hip_runtime.h>

#define D 64
#define LANES 16          // lanes per edge/node (4 floats each)
#define GROUPS 16         // edges (or nodes) per 256-thread block
#define LN_EPS 1e-5f

// ---------- CDNA5 async global->LDS staging (gfx1250) ----------
#if defined(__has_builtin)
# if __has_builtin(__builtin_amdgcn_global_load_async_to_lds_b128)
#  define USE_ASYNC_LDS 1
# endif
#endif

typedef int v4i __attribute__((vector_size(16)));
typedef v4i __attribute__((address_space(1))) gv4i;   // global
typedef v4i __attribute__((address_space(3))) lv4i;   // LDS

__device__ __forceinline__ void stage16(void* lds_dst, const void* gsrc) {
#if defined(USE_ASYNC_LDS)
  __builtin_amdgcn_global_load_async_to_lds_b128(
      (gv4i*)gsrc, (lv4i*)lds_dst, /*offset=*/0, /*cpol=*/0);
#else
  *(float4*)lds_dst = *(const float4*)gsrc;
#endif
}

__device__ __forceinline__ void stage_wait() {
#if defined(USE_ASYNC_LDS)
# if __has_builtin(__builtin_amdgcn_s_wait_asynccnt)
  __builtin_amdgcn_s_wait_asynccnt(0);
# else
  asm volatile("s_wait_asynccnt 0" ::: "memory");
# endif
#endif
  __syncthreads();
}

// ---------- small helpers ----------
__device__ __forceinline__ float4 ld4(const float* p) {
  return *reinterpret_cast<const float4*>(p);
}

// sum across the 16 lanes of this edge-group (wave32 -> two groups per wave)
__device__ __forceinline__ float rsum16(float x) {
  x += __shfl_xor(x, 1, 16);
  x += __shfl_xor(x, 2, 16);
  x += __shfl_xor(x, 4, 16);
  x += __shfl_xor(x, 8, 16);
  return x;
}

// order-preserving float->uint map for atomic max
__device__ __forceinline__ unsigned ordered_bits(float x) {
  unsigned b = __float_as_uint(x);
  return (b & 0x80000000u) ? ~b : (b | 0x80000000u);
}
__device__ __forceinline__ float from_ordered(unsigned u) {
  return (u & 0x80000000u) ? __uint_as_float(u ^ 0x80000000u)
                           : __uint_as_float(~u);
}

__device__ __forceinline__ void atomic_add_f32(float* p, float v) {
#if defined(__HIP_DEVICE_COMPILE__)
  unsafeAtomicAdd(p, v);   // lowers to global_atomic_add_f32
#else
  atomicAdd(p, v);
#endif
}

// elementwise edge-FFN + LayerNorm over D=64 (16 lanes x float4)
__device__ __forceinline__ float4 edge_ffn(
    float4 u, float4 v, float4 uw, float4 ub, float4 vw, float4 vb,
    float4 ew, float4 eb, float4 g, float4 b) {
  float4 h;
  h.x = fmaf(u.x, uw.x, ub.x) + fmaf(v.x, vw.x, vb.x);
  h.y = fmaf(u.y, uw.y, ub.y) + fmaf(v.y, vw.y, vb.y);
  h.z = fmaf(u.z, uw.z, ub.z) + fmaf(v.z, vw.z, vb.z);
  h.w = fmaf(u.w, uw.w, ub.w) + fmaf(v.w, vw.w, vb.w);
  h.x = fmaxf(h.x, 0.f); h.y = fmaxf(h.y, 0.f);
  h.z = fmaxf(h.z, 0.f); h.w = fmaxf(h.w, 0.f);
  h.x = fmaf(h.x, ew.x, eb.x);
  h.y = fmaf(h.y, ew.y, eb.y);
  h.z = fmaf(h.z, ew.z, eb.z);
  h.w = fmaf(h.w, ew.w, eb.w);
  float s  = h.x + h.y + h.z + h.w;
  float ss = h.x*h.x + h.y*h.y + h.z*h.z + h.w*h.w;
  s = rsum16(s); ss = rsum16(ss);
  float mu   = s  * (1.0f / D);
  float var  = ss * (1.0f / D) - mu * mu;
  float rstd = rsqrtf(fmaxf(var, 0.f) + LN_EPS);
  float4 o;
  o.x = fmaf((h.x - mu) * rstd, g.x, b.x);
  o.y = fmaf((h.y - mu) * rstd, g.y, b.y);
  o.z = fmaf((h.z - mu) * rstd, g.z, b.z);
  o.w = fmaf((h.w - mu) * rstd, g.w, b.w);
  return o;
}

// ---------- K0: zero accumulators ----------
__global__ __launch_bounds__(256) void k_init(float* __restrict__ out_accum,
                                              unsigned* __restrict__ seg_max,
                                              float* __restrict__ seg_sum,
                                              int n_nodes) {
  int i = blockIdx.x * 256 + threadIdx.x;
  if (i < n_nodes * D) out_accum[i] = 0.f;
  if (i < n_nodes) { seg_max[i] = 0u; seg_sum[i] = 0.f; }
}

// ---------- K1: key FFN -> score, per-dst running max ----------
__global__ __launch_bounds__(256) void k_edge_key(
    const float* __restrict__ feat, const float* __restrict__ query,
    const int* __restrict__ src, const int* __restrict__ dst,
    const float* __restrict__ w0, const float* __restrict__ w1,
    const float* __restrict__ w2, const float* __restrict__ w3,
    const float* __restrict__ w4, const float* __restrict__ w5,
    const float* __restrict__ gamma, const float* __restrict__ beta,
    float* __restrict__ score, unsigned* __restrict__ seg_max, int n_edges) {
  __shared__ float sw[6][GROUPS * D];            // 24 KB staged weight tiles
  const int tid = threadIdx.x;
  const long long fofs = (long long)blockIdx.x * (GROUPS * D) + tid * 4;
  if (fofs + 4 <= (long long)n_edges * D) {
    stage16(&sw[0][tid * 4], w0 + fofs);
    stage16(&sw[1][tid * 4], w1 + fofs);
    stage16(&sw[2][tid * 4], w2 + fofs);
    stage16(&sw[3][tid * 4], w3 + fofs);
    stage16(&sw[4][tid * 4], w4 + fofs);
    stage16(&sw[5][tid * 4], w5 + fofs);
  }
  stage_wait();

  const int lane = tid & (LANES - 1);
  const int grp  = tid >> 4;
  const int e = blockIdx.x * GROUPS + grp;
  if (e >= n_edges) return;
  const int sN = src[e], dN = dst[e];
  const int lofs = grp * D + lane * 4;
  float4 u = ld4(feat  + (long long)sN * D + lane * 4);
  float4 v = ld4(feat  + (long long)dN * D + lane * 4);
  float4 q = ld4(query + (long long)dN * D + lane * 4);
  float4 key = edge_ffn(u, v,
      ld4(&sw[0][lofs]), ld4(&sw[1][lofs]),
      ld4(&sw[2][lofs]), ld4(&sw[3][lofs]),
      ld4(&sw[4][lofs]), ld4(&sw[5][lofs]),
      ld4(gamma + lane * 4), ld4(beta + lane * 4));
  float dot = key.x*q.x + key.y*q.y + key.z*q.z + key.w*q.w;
  dot = rsum16(dot);
  if (lane == 0) {
    score[e] = dot;
    atomicMax(&seg_max[dN], ordered_bits(dot));
  }
}

// ---------- K2: ex = exp(score - max[dst]); per-dst sum ----------
__global__ __launch_bounds__(256) void k_exp(
    const int* __restrict__ dst, const unsigned* __restrict__ seg_max,
    float* __restrict__ score_ex, float* __restrict__ seg_sum, int n_edges) {
  int e = blockIdx.x * 256 + threadIdx.x;
  if (e >= n_edges) return;
  int dN = dst[e];
  float m = from_ordered(seg_max[dN]);
  float ex = __expf(score_ex[e] - m);
  score_ex[e] = ex;
  atomic_add_f32(&seg_sum[dN], ex);
}

// ---------- K3: value FFN, attn, scatter-add into node accumulator ----------
__global__ __launch_bounds__(256) void k_edge_val(
    const float* __restrict__ feat,
    const int* __restrict__ src, const int* __restrict__ dst,
    const float* __restrict__ w0, const float* __restrict__ w1,
    const float* __restrict__ w2, const float* __restrict__ w3,
    const float* __restrict__ w4, const float* __restrict__ w5,
    const float* __restrict__ gamma, const float* __restrict__ beta,
    const float* __restrict__ ex, const float* __restrict__ seg_sum,
    float* __restrict__ out_accum, float* __restrict__ attn_out, int n_edges) {
  __shared__ float sw[6][GROUPS * D];
  const int tid = threadIdx.x;
  const long long fofs = (long long)blockIdx.x * (GROUPS * D) + tid * 4;
  if (fofs + 4 <= (long long)n_edges * D) {
    stage16(&sw[0][tid * 4], w0 + fofs);
    stage16(&sw[1][tid * 4], w1 + fofs);
    stage16(&sw[2][tid * 4], w2 + fofs);
    stage16(&sw[3][tid * 4], w3 + fofs);
    stage16(&sw[4][tid * 4], w4 + fofs);
    stage16(&sw[5][tid * 4], w5 + fofs);
  }
  stage_wait();

  const int lane = tid & (LANES - 1);
  const int grp  = tid >> 4;
  const int e = blockIdx.x * GROUPS + grp;
  if (e >= n_edges) return;
  const int sN = src[e], dN = dst[e];
  const int lofs = grp * D + lane * 4;
  float attn = ex[e] / seg_sum[dN];
  float4 u = ld4(feat + (long long)sN * D + lane * 4);
  float4 v = ld4(feat + (long long)dN * D + lane * 4);
  float4 val = edge_ffn(u, v,
      ld4(&sw[0][lofs]), ld4(&sw[1][lofs]),
      ld4(&sw[2][lofs]), ld4(&sw[3][lofs]),
      ld4(&sw[4][lofs]), ld4(&sw[5][lofs]),
      ld4(gamma + lane * 4), ld4(beta + lane * 4));
  float* op = out_accum + (long long)dN * D + lane * 4;
  atomic_add_f32(op + 0, val.x * attn);
  atomic_add_f32(op + 1, val.y * attn);
  atomic_add_f32(op + 2, val.z * attn);
  atomic_add_f32(op + 3, val.w * attn);
  if (lane == 0) attn_out[e] = attn;
}

// ---------- K4: node LayerNorm (in place) ----------
__global__ __launch_bounds__(256) void k_node_ln(
    float* __restrict__ out, const float* __restrict__ gamma,
    const float* __restrict__ beta, int n_nodes) {
  const int tid = threadIdx.x;
  const int lane = tid & (LANES - 1);
  const int grp  = tid >> 4;
  const int n = blockIdx.x * GROUPS + grp;
  if (n >= n_nodes) return;
  float* p = out + (long long)n * D + lane * 4;
  float4 h = ld4(p);
  float s  = h.x + h.y + h.z + h.w;
  float ss = h.x*h.x + h.y*h.y + h.z*h.z + h.w*h.w;
  s = rsum16(s); ss = rsum16(ss);
  float mu   = s  * (1.0f / D);
  float var  = ss * (1.0f / D) - mu * mu;
  float rstd = rsqrtf(fmaxf(var, 0.f) + LN_EPS);
  float4 g = ld4(gamma + lane * 4), b = ld4(beta + lane * 4);
  float4 o;
  o.x = fmaf((h.x - mu) * rstd, g.x, b.x);
  o.y = fmaf((h.y - mu) * rstd, g.y, b.y);
  o.z = fmaf((h.z - mu) * rstd, g.z, b.z);
  o.w = fmaf((h.w - mu) * rstd, g.w, b.w);
  *(float4*)p = o;
}

extern "C" void kernel_launch(void* const* d_in, const int* in_sizes, int n_in,
                              void* d_out, int out_size, void* d_ws, size_t ws_size,
                              hipStream_t stream) {
  const float* feat  = (const float*)d_in[0];
  const float* query = (const float*)d_in[1];
  const int*   src   = (const int*)d_in[2];
  const int*   dst   = (const int*)d_in[3];
  const float* skw = (const float*)d_in[4];
  const float* skb = (const float*)d_in[5];
  const float* dkw = (const float*)d_in[6];
  const float* dkb = (const float*)d_in[7];
  const float* ekw = (const float*)d_in[8];
  const float* ekb = (const float*)d_in[9];
  const float* svw = (const float*)d_in[10];
  const float* svb = (const float*)d_in[11];
  const float* dvw = (const float*)d_in[12];
  const float* dvb = (const float*)d_in[13];
  const float* evw = (const float*)d_in[14];
  const float* evb = (const float*)d_in[15];
  const float* kg  = (const float*)d_in[16];
  const float* kb  = (const float*)d_in[17];
  const float* vg  = (const float*)d_in[18];
  const float* vb  = (const float*)d_in[19];
  const float* ag  = (const float*)d_in[20];
  const float* ab  = (const float*)d_in[21];

  const int nN = in_sizes[0] / D;   // 50000
  const int nE = in_sizes[2];       // 800000

  float*    out      = (float*)d_out;                 // [N*D]
  float*    attn_out = out + (long long)nN * D;       // [E]
  float*    ws_score = (float*)d_ws;                  // [E] score, then ex
  unsigned* seg_max  = (unsigned*)(ws_score + nE);    // [N] ordered bits
  float*    seg_sum  = (float*)(seg_max + nN);        // [N]

  dim3 blk(256);
  k_init<<<(nN * D + 255) / 256, blk, 0, stream>>>(out, seg_max, seg_sum, nN);
  k_edge_key<<<(nE + GROUPS - 1) / GROUPS, blk, 0, stream>>>(
      feat, query, src, dst, skw, skb, dkw, dkb, ekw, ekb, kg, kb,
      ws_score, seg_max, nE);
  k_exp<<<(nE + 255) / 256, blk, 0, stream>>>(dst, seg_max, ws_score, seg_sum, nE);
  k_edge_val<<<(nE + GROUPS - 1) / GROUPS, blk, 0, stream>>>(
      feat, src, dst, svw, svb, dvw, dvb, evw, evb, vg, vb,
      ws_score, seg_sum, out, attn_out, nE);
  k_node_ln<<<(nN + GROUPS - 1) / GROUPS, blk, 0, stream>>>(out, ag, ab, nN);
}